// StatNeighbor_79525614453056
// MI455X (gfx1250) — compile-verified
//
#include <hip/hip_runtime.h>

#define FDIM 128   // feature dim F
#define KDIM 256   // 2F (concat of x and src_fea)

typedef __attribute__((ext_vector_type(2))) float v2f;
typedef __attribute__((ext_vector_type(8))) float v8f;

// ---------------------------------------------------------------------------
// Kernel 1: zero the src_fea accumulator (N*F floats) in workspace
// ---------------------------------------------------------------------------
__global__ void zero_kernel(float4* __restrict__ ws, int n4) {
    int i = blockIdx.x * blockDim.x + threadIdx.x;
    if (i < n4) {
        float4 z; z.x = 0.f; z.y = 0.f; z.z = 0.f; z.w = 0.f;
        ws[i] = z;
    }
}

// ---------------------------------------------------------------------------
// Kernel 2: edge-parallel segment-sum scatter.
// One wave32 per edge; lane L owns floats [4L, 4L+4) of the 128-float row.
// x fits in the 192MB L2, so the random gather mostly hits L2; accumulation
// uses native global_atomic_add_f32 (unsafeAtomicAdd) into L2-resident ws.
// ---------------------------------------------------------------------------
__global__ void __launch_bounds__(256) scatter_kernel(
    const float* __restrict__ x,
    const int*   __restrict__ src,
    const int*   __restrict__ dst,
    float*       __restrict__ acc,
    int E)
{
    const int lane = threadIdx.x & 31;
    const int wavesPerBlock = blockDim.x >> 5;
    const int e = blockIdx.x * wavesPerBlock + (threadIdx.x >> 5);
    if (e >= E) return;

    const int s = src[e];
    const int d = dst[e];

    const float4 v = ((const float4*)(x + (size_t)s * FDIM))[lane];
    float* p = acc + (size_t)d * FDIM + lane * 4;
    unsafeAtomicAdd(p + 0, v.x);
    unsafeAtomicAdd(p + 1, v.y);
    unsafeAtomicAdd(p + 2, v.z);
    unsafeAtomicAdd(p + 3, v.w);
}

// ---------------------------------------------------------------------------
// Kernel 3: fused dual-GEMM + combine using V_WMMA_F32_16X16X4_F32.
//   in_fea  = [x | src_fea] @ W_in  + b_in
//   out_fea = [x | src_fea] @ W_out + b_out
//   out     = out_w * out_fea + in_w * in_fea
// One block (8 waves) per 16-node tile; wave w owns output cols [16w,16w+16).
// A layout (16x4 f32, ISA 7.12.2): lane<16 -> {K=k0,k0+1}, lane>=16 -> {k0+2,k0+3},
// row M = lane&15. B layout (4x16): lane<16 col=lane holds rows {k0,k0+1},
// lane>=16 col=lane-16 holds rows {k0+2,k0+3}.
// C/D layout: VGPR r -> M = r + (lane>=16 ? 8 : 0), N = lane&15.
// ---------------------------------------------------------------------------
__global__ void __launch_bounds__(256) gemm_combine_kernel(
    const float* __restrict__ x,
    const float* __restrict__ sfea,
    const float* __restrict__ in_w,
    const float* __restrict__ out_w,
    const float* __restrict__ W_in,
    const float* __restrict__ b_in,
    const float* __restrict__ W_out,
    const float* __restrict__ b_out,
    float* __restrict__ out)
{
    const int tile  = blockIdx.x;           // 16-node tile
    const int wave  = threadIdx.x >> 5;     // 0..7: output column tile
    const int lane  = threadIdx.x & 31;
    const int mrow  = lane & 15;            // A-matrix row for this lane
    const int khalf = (lane >> 4) << 1;     // 0 or 2: K sub-offset for this lane
    const int col   = (wave << 4) + (lane & 15);  // output column 0..127
    const int node0 = tile << 4;

    const float* xrow = x    + (size_t)(node0 + mrow) * FDIM;
    const float* srow = sfea + (size_t)(node0 + mrow) * FDIM;

    v8f acc_i = {};
    v8f acc_o = {};

    // K = 0..127 : A from x, W rows 0..127
    for (int k0 = 0; k0 < FDIM; k0 += 4) {
        const int kA = k0 + khalf;
        v2f a;  a.x  = xrow[kA];
                a.y  = xrow[kA + 1];
        v2f bi; bi.x = W_in [(size_t)kA * FDIM + col];
                bi.y = W_in [(size_t)(kA + 1) * FDIM + col];
        v2f bo; bo.x = W_out[(size_t)kA * FDIM + col];
                bo.y = W_out[(size_t)(kA + 1) * FDIM + col];
        acc_i = __builtin_amdgcn_wmma_f32_16x16x4_f32(false, a, false, bi,
                                                      (short)0, acc_i, false, false);
        acc_o = __builtin_amdgcn_wmma_f32_16x16x4_f32(false, a, false, bo,
                                                      (short)0, acc_o, false, false);
    }
    // K = 128..255 : A from src_fea, W rows 128..255
    for (int k0 = 0; k0 < FDIM; k0 += 4) {
        const int kA = k0 + khalf;
        const int kW = kA + FDIM;
        v2f a;  a.x  = srow[kA];
                a.y  = srow[kA + 1];
        v2f bi; bi.x = W_in [(size_t)kW * FDIM + col];
                bi.y = W_in [(size_t)(kW + 1) * FDIM + col];
        v2f bo; bo.x = W_out[(size_t)kW * FDIM + col];
                bo.y = W_out[(size_t)(kW + 1) * FDIM + col];
        acc_i = __builtin_amdgcn_wmma_f32_16x16x4_f32(false, a, false, bi,
                                                      (short)0, acc_i, false, false);
        acc_o = __builtin_amdgcn_wmma_f32_16x16x4_f32(false, a, false, bo,
                                                      (short)0, acc_o, false, false);
    }

    // Epilogue: out = out_w * (acc_o + b_out) + in_w * (acc_i + b_in)
    const int   rbase = (lane >> 4) << 3;   // 0 or 8
    const float bic   = b_in[col];
    const float boc   = b_out[col];
    #pragma unroll
    for (int r = 0; r < 8; ++r) {
        const int node = node0 + rbase + r;
        const float iw = in_w[node];
        const float ow = out_w[node];
        out[(size_t)node * FDIM + col] = ow * (acc_o[r] + boc) + iw * (acc_i[r] + bic);
    }
}

// ---------------------------------------------------------------------------
extern "C" void kernel_launch(void* const* d_in, const int* in_sizes, int n_in,
                              void* d_out, int out_size, void* d_ws, size_t ws_size,
                              hipStream_t stream) {
    const float* x     = (const float*)d_in[0];
    const int*   src   = (const int*)  d_in[1];
    const int*   dst   = (const int*)  d_in[2];
    const float* in_w  = (const float*)d_in[3];
    const float* out_w = (const float*)d_in[4];
    const float* W_in  = (const float*)d_in[5];
    const float* b_in  = (const float*)d_in[6];
    const float* W_out = (const float*)d_in[7];
    const float* b_out = (const float*)d_in[8];
    float*       out   = (float*)d_out;
    float*       ws    = (float*)d_ws;

    const int N = in_sizes[0] / FDIM;   // 100000
    const int E = in_sizes[1];          // 1600000

    // 1) zero src_fea accumulator
    const int n4 = (N * FDIM) / 4;
    zero_kernel<<<(n4 + 255) / 256, 256, 0, stream>>>((float4*)ws, n4);

    // 2) edge scatter (segment_sum) — 8 edges per 256-thread block
    const int eb = (E + 7) / 8;
    scatter_kernel<<<eb, 256, 0, stream>>>(x, src, dst, ws, E);

    // 3) fused dual-GEMM + combine — one block per 16-node tile (N % 16 == 0)
    gemm_combine_kernel<<<N / 16, 256, 0, stream>>>(x, ws, in_w, out_w,
                                                    W_in, b_in, W_out, b_out, out);
}